// TensorProductLinear_21303037788561
// MI455X (gfx1250) — compile-verified
//
#include <hip/hip_runtime.h>

// TensorProductLinear fused kernel for gfx1250 (MI455X), fp32 WMMA path.
//
//   o_l3[b,v,k] = (1/sqrt(tot)) * sum_paths sum_i A_p[b,v,i] * c_p[b,i,k]
//   A_p[b,v,i]  = sum_u x_p[b,u,i] * W_l3[wrow+u, v]          (WMMA f32 16x16x4, K=u)
//   c_p[b,i,k]  = sum_j cg_p[i,j,k] * y[b, yoff+j]            (precomputed per batch tile)
//
// Block = 256 threads (8 wave32), one 16-row batch tile per block; the 14
// output v-tiles (8 for l0, 4 for l1, 2 for l2) are distributed over waves.
// WMMA mapping: M = v (from W^T), N = batch, K = multiplicity u.
//
// Round-3 fixes: (a) D1>=3 K-loops kept rolled via an opaque trip count
// (full unroll pushed >256 VGPRs -> vgpr-msb juggling + mov storms);
// (b) W transposed into d_ws so A-fragments load as one global_load_b64
// (pairs (u,u+1) adjacent), with a direct-W fallback if ws is too small.

typedef float v2f __attribute__((ext_vector_type(2)));
typedef float v8f __attribute__((ext_vector_type(8)));

#define BATCH   16384
#define BT      16
#define NBT     (BATCH / BT)
#define XDIM    480
#define XPITCH  482      // even (8B-aligned b64 LDS loads), conflict-light
#define YPITCH  12
#define CROW    179      // floats of c per batch row
#define CPITCH  180
#define CG_TOTAL 615

// workspace layout (floats)
#define WS_CG    0
#define WS_WT0   640
#define WS_WT1   (WS_WT0 + 224*128)
#define WS_WT2   (WS_WT1 + 384*64)
#define WS_FLOATS (WS_WT2 + 352*32)

struct PathInfo {
  int l1, l2, l3;
  int d1, d2, d3;
  int mul1;
  int xoff, yoff;
  int wrow;
  int cgoff;
  int coff;
};

__constant__ PathInfo PATHS[15] = {
  {0,0,0, 1,1,1, 128,   0,0,   0,    0,   0},
  {1,1,0, 3,3,1,  64, 128,1, 128,    1,   1},
  {2,2,0, 5,5,1,  32, 320,4, 192,   10,   4},
  {0,1,1, 1,3,3, 128,   0,1,   0,   35,   9},
  {1,0,1, 3,1,3,  64, 128,0, 128,   44,  12},
  {1,1,1, 3,3,3,  64, 128,1, 192,   53,  21},
  {1,2,1, 3,5,3,  64, 128,4, 256,   80,  30},
  {2,1,1, 5,3,3,  32, 320,1, 320,  125,  39},
  {2,2,1, 5,5,3,  32, 320,4, 352,  170,  54},
  {0,2,2, 1,5,5, 128,   0,4,   0,  245,  69},
  {1,1,2, 3,3,5,  64, 128,1, 128,  270,  74},
  {1,2,2, 3,5,5,  64, 128,4, 192,  315,  89},
  {2,0,2, 5,1,5,  32, 320,0, 256,  390, 104},
  {2,1,2, 5,3,5,  32, 320,1, 288,  415, 129},
  {2,2,2, 5,5,5,  32, 320,4, 320,  490, 154},
};

// ---------------------------------------------------------------------------
// Init: real Clebsch-Gordan tensors (e3nn convention) in fp64 -> d_ws.
// ---------------------------------------------------------------------------
struct cplx { double re, im; };
__device__ inline cplx cmul(cplx a, cplx b){ return {a.re*b.re - a.im*b.im, a.re*b.im + a.im*b.re}; }
__device__ inline cplx conjc(cplx a){ return {a.re, -a.im}; }
__device__ inline double dfact(int n){ double r = 1.0; for (int i = 2; i <= n; ++i) r *= (double)i; return r; }

__device__ double su2_cg(int j1, int j2, int j3, int m1, int m2, int m3){
  if (m3 != m1 + m2) return 0.0;
  double pref = sqrt((2.0*j3+1.0)*dfact(j1+j2-j3)*dfact(j1-j2+j3)*dfact(-j1+j2+j3)/dfact(j1+j2+j3+1));
  pref *= sqrt(dfact(j3+m3)*dfact(j3-m3)*dfact(j1-m1)*dfact(j1+m1)*dfact(j2-m2)*dfact(j2+m2));
  double s = 0.0;
  for (int v = 0; v <= j1 + j2 - j3; ++v){
    int a = j1+j2-j3-v, b = j1-m1-v, c = j2+m2-v, d = j3-j2+m1+v, e = j3-j1-m2+v;
    if (a < 0 || b < 0 || c < 0 || d < 0 || e < 0) continue;
    double term = 1.0/(dfact(v)*dfact(a)*dfact(b)*dfact(c)*dfact(d)*dfact(e));
    s += (v & 1) ? -term : term;
  }
  return pref * s;
}

__device__ cplx qmat(int l, int r, int c){
  int m = r - l;
  const double inv = 0.70710678118654752440;
  cplx q = {0.0, 0.0};
  if (m < 0){
    if (c == l - m) q = { inv, 0.0};
    if (c == l + m) q = { 0.0, -inv};
  } else if (m == 0){
    if (c == l) q = {1.0, 0.0};
  } else {
    double sgn = (m & 1) ? -1.0 : 1.0;
    if (c == l + m) q = { sgn*inv, 0.0};
    if (c == l - m) q = { 0.0, sgn*inv};
  }
  cplx ph;
  switch (l & 3){ case 0: ph = {1,0}; break; case 1: ph = {0,-1}; break;
                  case 2: ph = {-1,0}; break; default: ph = {0,1}; break; }
  return cmul(ph, q);
}

__global__ void cg_init_kernel(float* __restrict__ ws){
  for (int e = threadIdx.x; e < CG_TOTAL; e += blockDim.x){
    int p = 0, rem = e;
    for (;;){
      int sz = PATHS[p].d1 * PATHS[p].d2 * PATHS[p].d3;
      if (rem < sz) break;
      rem -= sz; ++p;
    }
    const PathInfo P = PATHS[p];
    int i1 = rem / (P.d2 * P.d3);
    int i2 = (rem / P.d3) % P.d2;
    int i3 = rem % P.d3;
    double accr = 0.0;
    for (int a = 0; a < P.d1; ++a)
      for (int b = 0; b < P.d2; ++b)
        for (int c = 0; c < P.d3; ++c){
          double su2 = su2_cg(P.l1, P.l2, P.l3, a - P.l1, b - P.l2, c - P.l3);
          if (su2 == 0.0) continue;
          cplx t = cmul(qmat(P.l1, a, i1), qmat(P.l2, b, i2));
          t = cmul(t, conjc(qmat(P.l3, c, i3)));
          accr += t.re * su2;
        }
    ws[WS_CG + P.cgoff + rem] = (float)accr;
  }
}

// W[u][v] -> Wt[(u/2)*mulo + v][u&1] so (u, u+1) pairs are adjacent (b64 loads).
__global__ void w_transpose_kernel(const float* __restrict__ W0,
                                   const float* __restrict__ W1,
                                   const float* __restrict__ W2,
                                   float* __restrict__ ws){
  int idx = blockIdx.x * blockDim.x + threadIdx.x;
  if (idx < 224*128){ int u = idx >> 7, v = idx & 127;
    ws[WS_WT0 + (((u >> 1) << 7) + v)*2 + (u & 1)] = W0[idx]; }
  if (idx < 384*64){ int u = idx >> 6, v = idx & 63;
    ws[WS_WT1 + (((u >> 1) << 6) + v)*2 + (u & 1)] = W1[idx]; }
  if (idx < 352*32){ int u = idx >> 5, v = idx & 31;
    ws[WS_WT2 + (((u >> 1) << 5) + v)*2 + (u & 1)] = W2[idx]; }
}

// ---------------------------------------------------------------------------
// Main fused kernel
// ---------------------------------------------------------------------------
__device__ inline v8f zero8(){
  v8f z;
#pragma unroll
  for (int r = 0; r < 8; ++r) z[r] = 0.0f;
  return z;
}

__device__ inline void fma8(v8f& o, float s, const v8f& a){
#pragma unroll
  for (int r = 0; r < 8; ++r) o[r] += s * a[r];
}

__device__ inline int x_lds_col(int col){
  if (col < 128) return col;
  if (col < 320){ int t = col - 128; int u = t / 3; int i = t - u * 3;
                  return 128 + i * 64 + u; }
  int t = col - 320; int u = t / 5; int i = t - u * 5;
  return 320 + i * 32 + u;
}

// One path: GEMM over u with d1 accumulators, then CG*y fold into o[k].
// WT=true: W is the ws-transposed layout (A-frag = one b64).
template<int D1, int D3, bool WT>
__device__ inline void path_mm(v8f* o, const float* __restrict__ W, int mulo,
                               int mul1, int wrow, int xoff, int coff, int vb,
                               const float* s_x, const float* s_c, int lane)
{
  const int br   = lane & 15;
  const int klo  = (lane < 16) ? 0 : 2;   // K rows {0,2} in reg0, {1,3} in reg1
  const int vcol = vb + br;

  const float* wp;
  if (WT) wp = W + ((size_t)(((wrow + klo) >> 1) * mulo + vcol) << 1);
  else    wp = W + (size_t)(wrow + klo) * mulo + vcol;
  const float* xq = s_x + br * XPITCH + xoff + klo;

  v8f acc[D1];
  { // peeled first chunk: C is the inline-0 operand
    v2f a;
    if (WT) a = *(const v2f*)wp;
    else { a.x = wp[0]; a.y = wp[mulo]; }
#pragma unroll
    for (int i = 0; i < D1; ++i){
      v2f b = *(const v2f*)(xq + i * mul1);
      acc[i] = __builtin_amdgcn_wmma_f32_16x16x4_f32(
          false, a, false, b, (short)0, zero8(), false, false);
    }
    wp += 4 * mulo; xq += 4;
  }
  int nchunk = (mul1 >> 2) - 1;
  if (D1 > 1){
    // keep the multi-accumulator loops rolled: full unroll blew past 256 VGPRs
    asm volatile("" : "+s"(nchunk));
  }
  for (int uc = 0; uc < nchunk; ++uc){
    v2f a;
    if (WT) a = *(const v2f*)wp;
    else { a.x = wp[0]; a.y = wp[mulo]; }
#pragma unroll
    for (int i = 0; i < D1; ++i){
      v2f b = *(const v2f*)(xq + i * mul1);
      acc[i] = __builtin_amdgcn_wmma_f32_16x16x4_f32(
          false, a, false, b, (short)0, acc[i], false, false);
    }
    wp += 4 * mulo; xq += 4;
  }

  const float* cb = s_c + br * CPITCH + coff;
#pragma unroll
  for (int i = 0; i < D1; ++i)
#pragma unroll
    for (int k = 0; k < D3; ++k)
      fma8(o[k], cb[i * D3 + k], acc[i]);
}

template<int D3>
__device__ inline void store_out(const v8f* o, float* __restrict__ out,
                                 int b0, int colbase, int vb, float scale, int lane)
{
  const int br = lane & 15;
  const int hi = (lane >> 4) & 1;
  float* orow = out + (size_t)(b0 + br) * XDIM + colbase;
#pragma unroll
  for (int r = 0; r < 8; ++r){
    int v = vb + r + 8 * hi;
#pragma unroll
    for (int k = 0; k < D3; ++k)
      orow[v * D3 + k] = o[k][r] * scale;
  }
}

template<bool WT>
__global__ void __launch_bounds__(256)
tp_linear_kernel(const float* __restrict__ x, const float* __restrict__ y,
                 const float* __restrict__ W0g, const float* __restrict__ W1g,
                 const float* __restrict__ W2g, float* __restrict__ out,
                 const float* __restrict__ ws)
{
  __shared__ float s_x[BT * XPITCH];
  __shared__ float s_y[BT * YPITCH];
  __shared__ float s_c[BT * CPITCH];

  const int tid = threadIdx.x;
  const int b0  = blockIdx.x * BT;

  const float* cg = ws + WS_CG;
  const float* W0 = WT ? ws + WS_WT0 : W0g;
  const float* W1 = WT ? ws + WS_WT1 : W1g;
  const float* W2 = WT ? ws + WS_WT2 : W2g;

  // x tile -> LDS, transposed per l-block
  for (int idx = tid; idx < BT * (XDIM / 4); idx += 256){
    int b = idx / (XDIM / 4);
    int q = idx % (XDIM / 4);
    const float4 v = ((const float4*)(x + (size_t)(b0 + b) * XDIM))[q];
    float* dst = s_x + b * XPITCH;
    int col = q * 4;
    dst[x_lds_col(col    )] = v.x;
    dst[x_lds_col(col + 1)] = v.y;
    dst[x_lds_col(col + 2)] = v.z;
    dst[x_lds_col(col + 3)] = v.w;
  }
  for (int idx = tid; idx < BT * 9; idx += 256){
    int b = idx / 9, j = idx % 9;
    s_y[b * YPITCH + j] = y[(size_t)(b0 + b) * 9 + j];
  }
  __syncthreads();

  // c[b, p, i, k] = sum_j cg_p[i,j,k] * y[b, yoff+j]
  for (int idx = tid; idx < BT * CROW; idx += 256){
    int b = idx / CROW;
    int rem = idx % CROW;
    int p = 0;
    for (;;){
      int sz = PATHS[p].d1 * PATHS[p].d3;
      if (rem < sz) break;
      rem -= sz; ++p;
    }
    const PathInfo P = PATHS[p];
    int i = rem / P.d3;
    int k = rem % P.d3;
    float s = 0.0f;
    for (int j = 0; j < P.d2; ++j)
      s += cg[P.cgoff + (i * P.d2 + j) * P.d3 + k] * s_y[b * YPITCH + P.yoff + j];
    s_c[b * CPITCH + P.coff + rem] = s;
  }
  __syncthreads();

  const int wave = tid >> 5;
  const int lane = tid & 31;

  for (int vt = wave; vt < 14; vt += 8){
    if (vt < 8){
      const int vb = vt * 16;
      v8f o[1]; o[0] = zero8();
      path_mm<1,1,WT>(o, W0, 128, 128,   0,   0,   0, vb, s_x, s_c, lane);
      path_mm<3,1,WT>(o, W0, 128,  64, 128, 128,   1, vb, s_x, s_c, lane);
      path_mm<5,1,WT>(o, W0, 128,  32, 192, 320,   4, vb, s_x, s_c, lane);
      store_out<1>(o, out, b0, 0, vb, 0.066815310478106094f /*1/sqrt(224)*/, lane);
    } else if (vt < 12){
      const int vb = (vt - 8) * 16;
      v8f o[3];
#pragma unroll
      for (int k = 0; k < 3; ++k) o[k] = zero8();
      path_mm<1,3,WT>(o, W1, 64, 128,   0,   0,   9, vb, s_x, s_c, lane);
      path_mm<3,3,WT>(o, W1, 64,  64, 128, 128,  12, vb, s_x, s_c, lane);
      path_mm<3,3,WT>(o, W1, 64,  64, 192, 128,  21, vb, s_x, s_c, lane);
      path_mm<3,3,WT>(o, W1, 64,  64, 256, 128,  30, vb, s_x, s_c, lane);
      path_mm<5,3,WT>(o, W1, 64,  32, 320, 320,  39, vb, s_x, s_c, lane);
      path_mm<5,3,WT>(o, W1, 64,  32, 352, 320,  54, vb, s_x, s_c, lane);
      store_out<3>(o, out, b0, 128, vb, 0.051031036307982884f /*1/sqrt(384)*/, lane);
    } else {
      const int vb = (vt - 12) * 16;
      v8f o[5];
#pragma unroll
      for (int k = 0; k < 5; ++k) o[k] = zero8();
      path_mm<1,5,WT>(o, W2, 32, 128,   0,   0,  69, vb, s_x, s_c, lane);
      path_mm<3,5,WT>(o, W2, 32,  64, 128, 128,  74, vb, s_x, s_c, lane);
      path_mm<3,5,WT>(o, W2, 32,  64, 192, 128,  89, vb, s_x, s_c, lane);
      path_mm<5,5,WT>(o, W2, 32,  32, 256, 320, 104, vb, s_x, s_c, lane);
      path_mm<5,5,WT>(o, W2, 32,  32, 288, 320, 129, vb, s_x, s_c, lane);
      path_mm<5,5,WT>(o, W2, 32,  32, 320, 320, 154, vb, s_x, s_c, lane);
      store_out<5>(o, out, b0, 320, vb, 0.053300179088902614f /*1/sqrt(352)*/, lane);
    }
  }
}

extern "C" void kernel_launch(void* const* d_in, const int* in_sizes, int n_in,
                              void* d_out, int out_size, void* d_ws, size_t ws_size,
                              hipStream_t stream)
{
  const float* x  = (const float*)d_in[0];
  const float* y  = (const float*)d_in[1];
  const float* W0 = (const float*)d_in[2];
  const float* W1 = (const float*)d_in[3];
  const float* W2 = (const float*)d_in[4];
  float* out = (float*)d_out;
  float* ws  = (float*)d_ws;

  cg_init_kernel<<<1, 128, 0, stream>>>(ws);

  if (ws_size >= (size_t)WS_FLOATS * sizeof(float)){
    w_transpose_kernel<<<(224*128 + 255)/256, 256, 0, stream>>>(W0, W1, W2, ws);
    tp_linear_kernel<true><<<NBT, 256, 0, stream>>>(x, y, W0, W1, W2, out, ws);
  } else {
    tp_linear_kernel<false><<<NBT, 256, 0, stream>>>(x, y, W0, W1, W2, out, ws);
  }

  (void)in_sizes; (void)n_in; (void)out_size;
}